// MSLayer_42606075576368
// MI455X (gfx1250) — compile-verified
//
#include <hip/hip_runtime.h>
#include <stdint.h>

// LDPC min-sum layer: B=1024, N=4096, E=16384, colW=4, rowW=8.
// One workgroup per batch row; all gather tables live in LDS (80KB/WG).
constexpr int Nn    = 4096;
constexpr int Ee    = 16384;
constexpr int COLW  = 4;
constexpr int ROWM1 = 7;
constexpr int BLOCK = 1024;
constexpr int VPT   = Nn / BLOCK;  // 4 variable nodes per thread
constexpr int EPT   = Ee / BLOCK;  // 16 edges per thread

typedef unsigned int u32x4 __attribute__((ext_vector_type(4)));
typedef int          i32x8 __attribute__((ext_vector_type(8)));
typedef int          i32x4 __attribute__((ext_vector_type(4)));

__global__ __launch_bounds__(BLOCK) void msmin_kernel(
    const float* __restrict__ channelLLR,   // [B,N]
    const float* __restrict__ e2oLLR,       // [B,E]
    const int*   __restrict__ edgeToVar,    // [B,N,COLW]
    const float* __restrict__ edgeToVarMask,// [B,N,COLW]
    const int*   __restrict__ oddToEven,    // [B,E]
    const int*   __restrict__ edgeToChk,    // [B,E,ROWM1]
    const float* __restrict__ alpha,        // [1]
    float* __restrict__ outLLR,             // [B,N]
    float* __restrict__ outE2O)             // [B,E]
{
    __shared__ float lds_e[Ee];  // 64KB: e2oLLR row -> llr_edge (in place) -> e2o
    __shared__ float lds_v[Nn];  // 16KB: variable-node llr row

    const int b   = blockIdx.x;
    const int tid = threadIdx.x;

    const float* chRow = channelLLR    + (size_t)b * Nn;
    const float* eRow  = e2oLLR        + (size_t)b * Ee;
    const int*   evRow = edgeToVar     + (size_t)b * Nn * COLW;
    const float* emRow = edgeToVarMask + (size_t)b * Nn * COLW;
    const int*   oeRow = oddToEven     + (size_t)b * Ee;
    const int*   ecRow = edgeToChk     + (size_t)b * Ee * ROWM1;

    // ---- Phase 0: TDM async DMA of the contiguous 64KB e2oLLR row into LDS.
#if __has_builtin(__builtin_amdgcn_tensor_load_to_lds)
    if (tid < 32) {  // wave 0 issues one wave-level TDM op
        uint64_t ga = (uint64_t)(uintptr_t)eRow;
        uint32_t la = (uint32_t)(uintptr_t)(&lds_e[0]);  // low 32b of generic = LDS offset
        u32x4 g0;
        g0[0] = 1u;                                        // count=1, user descriptor
        g0[1] = la;                                        // lds_addr
        g0[2] = (uint32_t)(ga & 0xFFFFFFFFu);              // global_addr[31:0]
        g0[3] = (uint32_t)((ga >> 32) & 0x01FFFFFFu)       // global_addr[56:32]
              | (2u << 30);                                // type=2 ("image")
        i32x8 g1;
        g1[0] = (int)(2u << 16);                           // wg_mask=0, data_size=4B
        g1[1] = (int)((uint32_t)(Ee & 0xFFFF) << 16);      // tensor_dim0 lo16
        g1[2] = (int)(((uint32_t)Ee >> 16) | (1u << 16));  // tensor_dim0 hi16 | tensor_dim1=1
        g1[3] = (int)((uint32_t)(Ee & 0xFFFF) << 16);      // tensor_dim1 hi=0 | tile_dim0=E
        g1[4] = 1;                                         // tile_dim1=1, tile_dim2=0
        g1[5] = Ee;                                        // tensor_dim0_stride lo32
        g1[6] = 0;                                         // stride0 hi16 | stride1 lo16
        g1[7] = 0;                                         // stride1 hi32
        i32x4 gz4 = {0, 0, 0, 0};
        i32x8 gz8 = {0, 0, 0, 0, 0, 0, 0, 0};
        // amdgpu-toolchain (clang-23) 6-arg form: extra int32x8 descriptor group.
        __builtin_amdgcn_tensor_load_to_lds(g0, g1, gz4, gz4, gz8, 0);
    }
#else
    for (int i = tid; i < Ee; i += BLOCK) lds_e[i] = eRow[i];
#endif

    // While the DMA flies: pull this thread's per-variable-node data into regs.
    int   vidx[VPT][COLW];
    float vmsk[VPT][COLW];
    float vch[VPT];
#pragma unroll
    for (int i = 0; i < VPT; ++i) {
        const int n = tid + i * BLOCK;
        vch[i] = chRow[n];
        int4   iv = *(const int4*)  (evRow + (size_t)n * COLW);  // 16B aligned
        float4 mv = *(const float4*)(emRow + (size_t)n * COLW);
        vidx[i][0] = iv.x; vidx[i][1] = iv.y; vidx[i][2] = iv.z; vidx[i][3] = iv.w;
        vmsk[i][0] = mv.x; vmsk[i][1] = mv.y; vmsk[i][2] = mv.z; vmsk[i][3] = mv.w;
    }
    const float av = alpha[0];

#if __has_builtin(__builtin_amdgcn_s_wait_tensorcnt)
    __builtin_amdgcn_s_wait_tensorcnt(0);
#else
    asm volatile("s_wait_tensorcnt 0" ::: "memory");
#endif
    __syncthreads();

    // ---- Phase 1: variable-node update. llr[n] = ch + sum_k e2o[idx]*mask
#pragma unroll
    for (int i = 0; i < VPT; ++i) {
        const int n = tid + i * BLOCK;
        float s = vch[i];
#pragma unroll
        for (int k = 0; k < COLW; ++k) s += lds_e[vidx[i][k]] * vmsk[i][k];
        lds_v[n] = s;
    }
    __syncthreads();  // publishes lds_v; also fences phase-1 reads of lds_e

    // ---- Phase 2: extrinsic edge messages, in place: lds_e[e] = llr[o2e[e]] - e2o_in[e]
#pragma unroll
    for (int i = 0; i < EPT; ++i) {
        const int e = tid + i * BLOCK;
        const int o = oeRow[e];
        __builtin_prefetch(ecRow + (size_t)e * ROWM1, 0, 1);  // global_prefetch for phase 3
        lds_e[e] = lds_v[o] - lds_e[e];  // each thread owns its e: no cross-thread hazard
    }
    __syncthreads();

    // ---- Phase 3: check-node min-sum over 7 peer edges (gathers hit LDS)
    float r[EPT];
#pragma unroll
    for (int i = 0; i < EPT; ++i) {
        const int e = tid + i * BLOCK;
        const int* p = ecRow + (size_t)e * ROWM1;
        float    mn  = 3.402823466e+38f;
        uint32_t sgn = 0u;
#pragma unroll
        for (int j = 0; j < ROWM1; ++j) {
            const uint32_t bits = __float_as_uint(lds_e[p[j]]);
            sgn ^= (bits & 0x80000000u);
            mn = fminf(mn, __uint_as_float(bits & 0x7FFFFFFFu));
        }
        r[i] = __uint_as_float(__float_as_uint(mn * av) ^ sgn);
        outE2O[(size_t)b * Ee + e] = r[i];  // coalesced store of e2o output
    }
    __syncthreads();  // all random reads of lds_e (llr_edge) done

    // ---- Phase 3b: publish e2o into LDS for the final marginalization
#pragma unroll
    for (int i = 0; i < EPT; ++i) lds_e[tid + i * BLOCK] = r[i];
    __syncthreads();

    // ---- Phase 4: posterior LLR per variable node (reuse cached idx/mask/ch regs)
#pragma unroll
    for (int i = 0; i < VPT; ++i) {
        const int n = tid + i * BLOCK;
        float s = vch[i];
#pragma unroll
        for (int k = 0; k < COLW; ++k) s += lds_e[vidx[i][k]] * vmsk[i][k];
        outLLR[(size_t)b * Nn + n] = s;
    }
}

extern "C" void kernel_launch(void* const* d_in, const int* in_sizes, int n_in,
                              void* d_out, int out_size, void* d_ws, size_t ws_size,
                              hipStream_t stream) {
    (void)n_in; (void)out_size; (void)d_ws; (void)ws_size;
    const float* channelLLR = (const float*)d_in[0];
    const float* e2oLLR     = (const float*)d_in[1];
    const int*   edgeToVar  = (const int*)  d_in[3];
    const float* mask       = (const float*)d_in[4];
    const int*   oddToEven  = (const int*)  d_in[5];
    const int*   edgeToChk  = (const int*)  d_in[6];
    const float* alpha      = (const float*)d_in[8];

    const int B = in_sizes[0] / Nn;  // channelLLR is [B,N]
    float* outLLR = (float*)d_out;
    float* outE2O = outLLR + (size_t)B * Nn;

    msmin_kernel<<<B, BLOCK, 0, stream>>>(channelLLR, e2oLLR, edgeToVar, mask,
                                          oddToEven, edgeToChk, alpha,
                                          outLLR, outE2O);
}